// AttentionFlow_13915694039645
// MI455X (gfx1250) — compile-verified
//
#include <hip/hip_runtime.h>
#include <math.h>

typedef __attribute__((ext_vector_type(2))) float v2f;
typedef __attribute__((ext_vector_type(8))) float v8f;

#define E_CONST 250000
#define B_CONST 16
#define N_CONST 50000
#define R_CONST 500
#define DIN     100
#define DD      64
#define M_CONST 500000

static __device__ inline v8f wmma4(v2f a, v2f b, v8f c) {
  return __builtin_amdgcn_wmma_f32_16x16x4_f32(false, a, false, b, (short)0, c, false, false);
}

static __device__ inline float lrelu(float x) { return x > 0.f ? x : 0.2f * x; }

// monotonic float->uint key for atomic max
static __device__ inline unsigned fkey(float f) {
  unsigned u = __float_as_uint(f);
  return (u & 0x80000000u) ? ~u : (u | 0x80000000u);
}
static __device__ inline float funkey(unsigned k) {
  return (k & 0x80000000u) ? __uint_as_float(k ^ 0x80000000u) : __uint_as_float(~k);
}

// ---------------------------------------------------------------------------
// Y[r, 0:64] = tanh(X[r, 0:100] @ W(100,64) + b)   (WMMA f32 16x16x4)
// one 16-row strip per wave, 8 waves / block (128 rows per block)
// ---------------------------------------------------------------------------
__global__ void __launch_bounds__(256) proj_tanh_kernel(
    const float* __restrict__ X, const float* __restrict__ W,
    const float* __restrict__ b, float* __restrict__ Y, int Rn) {
  const int wave = threadIdx.x >> 5;
  const int lane = threadIdx.x & 31;
  const int half = lane >> 4;
  const int li   = lane & 15;
  long rowbase = ((long)blockIdx.x * 8 + wave) * 16;
  if (rowbase >= Rn) return;  // wave-uniform
  long myrow = rowbase + li;
  if (myrow >= Rn) myrow = Rn - 1;
  const float* xr = X + myrow * DIN;

  v8f acc[4];
#pragma unroll
  for (int j = 0; j < 4; ++j)
#pragma unroll
    for (int v = 0; v < 8; ++v) acc[j][v] = 0.f;

#pragma unroll
  for (int k4 = 0; k4 < DIN / 4; ++k4) {
    const int kk = 4 * k4 + 2 * half;
    v2f a;
    a.x = xr[kk];
    a.y = xr[kk + 1];
#pragma unroll
    for (int j = 0; j < 4; ++j) {
      v2f bb;
      bb.x = W[kk * DD + j * 16 + li];
      bb.y = W[(kk + 1) * DD + j * 16 + li];
      acc[j] = wmma4(a, bb, acc[j]);
    }
  }
#pragma unroll
  for (int j = 0; j < 4; ++j) {
    const int n = j * 16 + li;
    const float bias = b[n];
#pragma unroll
    for (int v = 0; v < 8; ++v) {
      long row = rowbase + v + 8 * half;
      if (row < Rn) Y[row * DD + n] = tanhf(acc[j][v] + bias);
    }
  }
}

// ---------------------------------------------------------------------------
// fold rel / query tables (tiny)
// ---------------------------------------------------------------------------
__global__ void tables_kernel(
    const float* __restrict__ remb, const float* __restrict__ qh,
    const float* __restrict__ qr,
    const float* __restrict__ g1lW, const float* __restrict__ g1lb,
    const float* __restrict__ g1rW, const float* __restrict__ g1rb,
    const float* __restrict__ g2lW, const float* __restrict__ g2lb,
    const float* __restrict__ g2rW, const float* __restrict__ g2rb,
    float* __restrict__ rl1, float* __restrict__ rr1,
    float* __restrict__ rl2, float* __restrict__ rr2,
    float* __restrict__ ql1, float* __restrict__ qr1,
    float* __restrict__ ql2, float* __restrict__ qr2) {
  int idx = blockIdx.x * blockDim.x + threadIdx.x;
  const int total_rel = 4 * R_CONST * DD;
  if (idx < total_rel) {
    int t = idx / (R_CONST * DD);
    int r = (idx / DD) % R_CONST;
    int n = idx & (DD - 1);
    const float* W = (t == 0) ? g1lW : (t == 1) ? g1rW : (t == 2) ? g2lW : g2rW;
    float s = 0.f;
    for (int k = 0; k < DD; ++k) s += remb[r * DD + k] * W[(64 + k) * DD + n];
    float* T = (t == 0) ? rl1 : (t == 1) ? rr1 : (t == 2) ? rl2 : rr2;
    T[r * DD + n] = s;
  } else {
    int id2 = idx - total_rel;
    if (id2 >= 4 * B_CONST * DD) return;
    int t = id2 / (B_CONST * DD);
    int bi = (id2 / DD) % B_CONST;
    int n = id2 & (DD - 1);
    const float* W;
    const float* bia;
    float* T;
    if (t == 0)      { W = g1lW; bia = g1lb; T = ql1; }
    else if (t == 1) { W = g1rW; bia = g1rb; T = qr1; }
    else if (t == 2) { W = g2lW; bia = g2lb; T = ql2; }
    else             { W = g2rW; bia = g2rb; T = qr2; }
    float s = bia[n];
    for (int k = 0; k < DD; ++k) {
      s += qh[bi * DD + k] * W[(128 + k) * DD + n];
      s += qr[bi * DD + k] * W[(192 + k) * DD + n];
    }
    T[bi * DD + n] = s;
  }
}

// hu2 = hu_proj @ g2_rW[0:64,:]   (N x 64, small)
__global__ void hu2_kernel(const float* __restrict__ hu_proj,
                           const float* __restrict__ g2rW,
                           float* __restrict__ hu2) {
  long idx = (long)blockIdx.x * blockDim.x + threadIdx.x;
  if (idx >= (long)N_CONST * DD) return;
  int r = (int)(idx >> 6);
  int n = (int)(idx & (DD - 1));
  float s = 0.f;
  for (int k = 0; k < DD; ++k) s += hu_proj[(long)r * DD + k] * g2rW[k * DD + n];
  hu2[idx] = s;
}

// ---------------------------------------------------------------------------
// per-edge gates: 16 edges per wave, WMMA f32 16x16x4 chains
// ---------------------------------------------------------------------------
__global__ void __launch_bounds__(256) edge_logits_kernel(
    const int* __restrict__ edges,
    const float* __restrict__ hp, const float* __restrict__ hu2,
    const float* __restrict__ rl1, const float* __restrict__ rr1,
    const float* __restrict__ rl2, const float* __restrict__ rr2,
    const float* __restrict__ ql1, const float* __restrict__ qr1,
    const float* __restrict__ ql2, const float* __restrict__ qr2,
    const float* __restrict__ w_l1, const float* __restrict__ w_r1,
    const float* __restrict__ w_c1, const float* __restrict__ b_c1,
    const float* __restrict__ w_l2, const float* __restrict__ w_c2,
    const float* __restrict__ b_c2, float* __restrict__ logits) {
  __shared__ float ldsr[8 * 16 * 68];
  const int wave = threadIdx.x >> 5;
  const int lane = threadIdx.x & 31;
  const int half = lane >> 4;
  const int li   = lane & 15;
  float* Lr = &ldsr[wave * 16 * 68];

  long ebase = ((long)blockIdx.x * 8 + wave) * 16;
  long eme = ebase + li;
  if (eme >= E_CONST) eme = E_CONST - 1;
  const int eg  = edges[eme * 8 + 0];
  const int vj  = edges[eme * 8 + 2];
  const int rel = edges[eme * 8 + 3];
  const int i6  = edges[eme * 8 + 6];
  const int i7  = edges[eme * 8 + 7];

  // per-row broadcast scalars (row M = v + 8*half)
  int relm8[8], egm8[8], vjm8[8];
#pragma unroll
  for (int v = 0; v < 8; ++v) {
    int Mrow = v + 8 * half;
    relm8[v] = __shfl(rel, Mrow, 32);
    egm8[v]  = __shfl(eg, Mrow, 32);
    vjm8[v]  = __shfl(vj, Mrow, 32);
  }

  v2f a[16];
  v8f l1r[4], l2r[4];
  float lacc[8];
#pragma unroll
  for (int v = 0; v < 8; ++v) lacc[v] = 0.f;

  // ----- A = hp[e6] rows ------------------------------------------------
  {
    const float* hr = hp + (long)i6 * DD;
#pragma unroll
    for (int k = 0; k < 16; ++k) {
      int kk = 4 * k + 2 * half;
      a[k].x = hr[kk];
      a[k].y = hr[kk + 1];
    }
  }
  // l1 = lrelu(Hvi @ w_l1[0:64] + rl1[rel] + ql1[eg])  (kept in regs, C-layout)
#pragma unroll
  for (int j = 0; j < 4; ++j) {
    v8f acc;
#pragma unroll
    for (int v = 0; v < 8; ++v) acc[v] = 0.f;
    const int n = j * 16 + li;
#pragma unroll
    for (int k = 0; k < 16; ++k) {
      int kk = 4 * k + 2 * half;
      v2f bb;
      bb.x = w_l1[kk * DD + n];
      bb.y = w_l1[(kk + 1) * DD + n];
      acc = wmma4(a[k], bb, acc);
    }
#pragma unroll
    for (int v = 0; v < 8; ++v)
      acc[v] = lrelu(acc[v] + rl1[relm8[v] * DD + n] + ql1[egm8[v] * DD + n]);
    l1r[j] = acc;
  }
  // l2 (same A, g2 weights)
#pragma unroll
  for (int j = 0; j < 4; ++j) {
    v8f acc;
#pragma unroll
    for (int v = 0; v < 8; ++v) acc[v] = 0.f;
    const int n = j * 16 + li;
#pragma unroll
    for (int k = 0; k < 16; ++k) {
      int kk = 4 * k + 2 * half;
      v2f bb;
      bb.x = w_l2[kk * DD + n];
      bb.y = w_l2[(kk + 1) * DD + n];
      acc = wmma4(a[k], bb, acc);
    }
#pragma unroll
    for (int v = 0; v < 8; ++v)
      acc[v] = lrelu(acc[v] + rl2[relm8[v] * DD + n] + ql2[egm8[v] * DD + n]);
    l2r[j] = acc;
  }
  // ----- r1 = lrelu(Hvj @ w_r1[0:64] + rr1 + qr1) -> LDS (transpose) ----
  {
    const float* hr = hp + (long)i7 * DD;
#pragma unroll
    for (int k = 0; k < 16; ++k) {
      int kk = 4 * k + 2 * half;
      a[k].x = hr[kk];
      a[k].y = hr[kk + 1];
    }
  }
#pragma unroll
  for (int j = 0; j < 4; ++j) {
    v8f acc;
#pragma unroll
    for (int v = 0; v < 8; ++v) acc[v] = 0.f;
    const int n = j * 16 + li;
#pragma unroll
    for (int k = 0; k < 16; ++k) {
      int kk = 4 * k + 2 * half;
      v2f bb;
      bb.x = w_r1[kk * DD + n];
      bb.y = w_r1[(kk + 1) * DD + n];
      acc = wmma4(a[k], bb, acc);
    }
#pragma unroll
    for (int v = 0; v < 8; ++v) {
      int Mrow = v + 8 * half;
      Lr[Mrow * 68 + n] =
          lrelu(acc[v] + rr1[relm8[v] * DD + n] + qr1[egm8[v] * DD + n]);
    }
  }
  __syncthreads();
  // reload r1 as A-matrix (row per lane) from LDS
#pragma unroll
  for (int k = 0; k < 16; ++k) {
    int kk = 4 * k + 2 * half;
    a[k].x = Lr[li * 68 + kk];
    a[k].y = Lr[li * 68 + kk + 1];
  }
  __syncthreads();
  // c1 = r1 @ w_c1 + b_c1 ; lacc += c1 * l1
#pragma unroll
  for (int j = 0; j < 4; ++j) {
    v8f acc;
#pragma unroll
    for (int v = 0; v < 8; ++v) acc[v] = 0.f;
    const int n = j * 16 + li;
#pragma unroll
    for (int k = 0; k < 16; ++k) {
      int kk = 4 * k + 2 * half;
      v2f bb;
      bb.x = w_c1[kk * DD + n];
      bb.y = w_c1[(kk + 1) * DD + n];
      acc = wmma4(a[k], bb, acc);
    }
    const float cb = b_c1[n];
#pragma unroll
    for (int v = 0; v < 8; ++v) lacc[v] += (acc[v] + cb) * l1r[j][v];
  }
  // ----- r2 = lrelu(hu2[vj] + rr2 + qr2) -> LDS (no matmul needed) ------
#pragma unroll
  for (int v = 0; v < 8; ++v) {
    int Mrow = v + 8 * half;
#pragma unroll
    for (int j = 0; j < 4; ++j) {
      int n = j * 16 + li;
      float x = hu2[(long)vjm8[v] * DD + n] + rr2[relm8[v] * DD + n] +
                qr2[egm8[v] * DD + n];
      Lr[Mrow * 68 + n] = lrelu(x);
    }
  }
  __syncthreads();
#pragma unroll
  for (int k = 0; k < 16; ++k) {
    int kk = 4 * k + 2 * half;
    a[k].x = Lr[li * 68 + kk];
    a[k].y = Lr[li * 68 + kk + 1];
  }
  // c2 = r2 @ w_c2 + b_c2 ; lacc += c2 * l2
#pragma unroll
  for (int j = 0; j < 4; ++j) {
    v8f acc;
#pragma unroll
    for (int v = 0; v < 8; ++v) acc[v] = 0.f;
    const int n = j * 16 + li;
#pragma unroll
    for (int k = 0; k < 16; ++k) {
      int kk = 4 * k + 2 * half;
      v2f bb;
      bb.x = w_c2[kk * DD + n];
      bb.y = w_c2[(kk + 1) * DD + n];
      acc = wmma4(a[k], bb, acc);
    }
    const float cb = b_c2[n];
#pragma unroll
    for (int v = 0; v < 8; ++v) lacc[v] += (acc[v] + cb) * l2r[j][v];
  }
  // ----- reduce over N (16 lanes per half) and store logits -------------
#pragma unroll
  for (int v = 0; v < 8; ++v) {
    float s = lacc[v];
    s += __shfl_xor(s, 1, 32);
    s += __shfl_xor(s, 2, 32);
    s += __shfl_xor(s, 4, 32);
    s += __shfl_xor(s, 8, 32);
    if (li == 0) {
      long e = ebase + v + 8 * half;
      if (e < E_CONST) logits[e] = s;
    }
  }
}

// ---------------------------------------------------------------------------
// segmented softmax + scatter + normalize
// ---------------------------------------------------------------------------
__global__ void seg_init_kernel(unsigned* __restrict__ smax,
                                float* __restrict__ den,
                                float* __restrict__ naw) {
  long i = (long)blockIdx.x * blockDim.x + threadIdx.x;
  if (i < E_CONST) {
    smax[i] = 0u;
    den[i] = 0.f;
  }
  if (i < (long)B_CONST * N_CONST) naw[i] = 0.f;
}

__global__ void seg_max_kernel(const int* __restrict__ edges,
                               const float* __restrict__ logits,
                               unsigned* __restrict__ smax) {
  long e = (long)blockIdx.x * blockDim.x + threadIdx.x;
  if (e >= E_CONST) return;
  int seg = edges[e * 8 + 4];
  atomicMax(&smax[seg], fkey(logits[e]));
}

__global__ void seg_exp_kernel(const int* __restrict__ edges,
                               const float* __restrict__ logits,
                               const unsigned* __restrict__ smax,
                               float* __restrict__ ex, float* __restrict__ den) {
  long e = (long)blockIdx.x * blockDim.x + threadIdx.x;
  if (e >= E_CONST) return;
  int seg = edges[e * 8 + 4];
  float x = expf(logits[e] - funkey(smax[seg]));
  ex[e] = x;
  atomicAdd(&den[seg], x);
}

__global__ void scatter_kernel(const int* __restrict__ edges,
                               const float* __restrict__ ex,
                               const float* __restrict__ den,
                               const float* __restrict__ na,
                               float* __restrict__ naw) {
  long e = (long)blockIdx.x * blockDim.x + threadIdx.x;
  if (e >= E_CONST) return;
  int eg = edges[e * 8 + 0];
  int vi = edges[e * 8 + 1];
  int vj = edges[e * 8 + 2];
  int seg = edges[e * 8 + 4];
  float t = ex[e] / den[seg];
  float ta = na[(long)eg * N_CONST + vi] * t;
  atomicAdd(&naw[(long)eg * N_CONST + vj], ta);
}

__global__ void __launch_bounds__(512) norm_kernel(const float* __restrict__ naw,
                                                   float* __restrict__ out) {
  __shared__ float red[512];
  int b = blockIdx.x;
  float s = 0.f;
  for (int i = threadIdx.x; i < N_CONST; i += 512)
    s += naw[(long)b * N_CONST + i];
  red[threadIdx.x] = s;
  __syncthreads();
  for (int st = 256; st > 0; st >>= 1) {
    if ((int)threadIdx.x < st) red[threadIdx.x] += red[threadIdx.x + st];
    __syncthreads();
  }
  float inv = 1.0f / red[0];
  for (int i = threadIdx.x; i < N_CONST; i += 512)
    out[(long)b * N_CONST + i] = naw[(long)b * N_CONST + i] * inv;
}

// ---------------------------------------------------------------------------
extern "C" void kernel_launch(void* const* d_in, const int* in_sizes, int n_in,
                              void* d_out, int out_size, void* d_ws,
                              size_t ws_size, hipStream_t stream) {
  const float* na     = (const float*)d_in[0];
  const int*   edges  = (const int*)d_in[1];
  const float* h_unc  = (const float*)d_in[2];
  const float* h_con  = (const float*)d_in[3];
  const float* rel_t  = (const float*)d_in[4];
  const float* qh_emb = (const float*)d_in[5];
  const float* qr_emb = (const float*)d_in[6];
  const float* p_W    = (const float*)d_in[7];
  const float* p_b    = (const float*)d_in[8];
  const float* g1lW = (const float*)d_in[9];
  const float* g1lb = (const float*)d_in[10];
  const float* g1rW = (const float*)d_in[11];
  const float* g1rb = (const float*)d_in[12];
  const float* g1cW = (const float*)d_in[13];
  const float* g1cb = (const float*)d_in[14];
  const float* g2lW = (const float*)d_in[15];
  const float* g2lb = (const float*)d_in[16];
  const float* g2rW = (const float*)d_in[17];
  const float* g2rb = (const float*)d_in[18];
  const float* g2cW = (const float*)d_in[19];
  const float* g2cb = (const float*)d_in[20];
  float* out = (float*)d_out;

  float* ws = (float*)d_ws;
  const long OFF_HP   = 0;
  const long OFF_HU   = OFF_HP + (long)M_CONST * DD;
  const long OFF_HU2  = OFF_HU + (long)N_CONST * DD;
  const long OFF_REMB = OFF_HU2 + (long)N_CONST * DD;
  const long OFF_QHP  = OFF_REMB + (long)R_CONST * DD;
  const long OFF_QRP  = OFF_QHP + (long)B_CONST * DD;
  const long OFF_RL1  = OFF_QRP + (long)B_CONST * DD;
  const long OFF_RR1  = OFF_RL1 + (long)R_CONST * DD;
  const long OFF_RL2  = OFF_RR1 + (long)R_CONST * DD;
  const long OFF_RR2  = OFF_RL2 + (long)R_CONST * DD;
  const long OFF_QL1  = OFF_RR2 + (long)R_CONST * DD;
  const long OFF_QR1  = OFF_QL1 + (long)B_CONST * DD;
  const long OFF_QL2  = OFF_QR1 + (long)B_CONST * DD;
  const long OFF_QR2  = OFF_QL2 + (long)B_CONST * DD;
  const long OFF_LOG  = OFF_QR2 + (long)B_CONST * DD;
  const long OFF_SMX  = OFF_LOG + E_CONST;
  const long OFF_DEN  = OFF_SMX + E_CONST;
  const long OFF_EX   = OFF_DEN + E_CONST;
  const long OFF_NA   = OFF_EX + E_CONST;

  // Phase A: projections (WMMA)
  proj_tanh_kernel<<<(M_CONST + 127) / 128, 256, 0, stream>>>(
      h_con, p_W, p_b, ws + OFF_HP, M_CONST);
  proj_tanh_kernel<<<(N_CONST + 127) / 128, 256, 0, stream>>>(
      h_unc, p_W, p_b, ws + OFF_HU, N_CONST);
  proj_tanh_kernel<<<(R_CONST + 127) / 128, 256, 0, stream>>>(
      rel_t, p_W, p_b, ws + OFF_REMB, R_CONST);
  proj_tanh_kernel<<<1, 256, 0, stream>>>(qh_emb, p_W, p_b, ws + OFF_QHP,
                                          B_CONST);
  proj_tanh_kernel<<<1, 256, 0, stream>>>(qr_emb, p_W, p_b, ws + OFF_QRP,
                                          B_CONST);
  // fold rel / query tables
  {
    int tot = 4 * R_CONST * DD + 4 * B_CONST * DD;
    tables_kernel<<<(tot + 255) / 256, 256, 0, stream>>>(
        ws + OFF_REMB, ws + OFF_QHP, ws + OFF_QRP, g1lW, g1lb, g1rW, g1rb,
        g2lW, g2lb, g2rW, g2rb, ws + OFF_RL1, ws + OFF_RR1, ws + OFF_RL2,
        ws + OFF_RR2, ws + OFF_QL1, ws + OFF_QR1, ws + OFF_QL2, ws + OFF_QR2);
  }
  hu2_kernel<<<(int)(((long)N_CONST * DD + 255) / 256), 256, 0, stream>>>(
      ws + OFF_HU, g2rW, ws + OFF_HU2);

  // Phase B: per-edge gated logits (WMMA)
  edge_logits_kernel<<<(E_CONST + 127) / 128, 256, 0, stream>>>(
      edges, ws + OFF_HP, ws + OFF_HU2, ws + OFF_RL1, ws + OFF_RR1,
      ws + OFF_RL2, ws + OFF_RR2, ws + OFF_QL1, ws + OFF_QR1, ws + OFF_QL2,
      ws + OFF_QR2, g1lW, g1rW, g1cW, g1cb, g2lW, g2cW, g2cb, ws + OFF_LOG);

  // Phase C: segmented softmax + scatter + normalize
  {
    long initn = (long)B_CONST * N_CONST;  // >= E_CONST
    seg_init_kernel<<<(int)((initn + 255) / 256), 256, 0, stream>>>(
        (unsigned*)(ws + OFF_SMX), ws + OFF_DEN, ws + OFF_NA);
  }
  seg_max_kernel<<<(E_CONST + 255) / 256, 256, 0, stream>>>(
      edges, ws + OFF_LOG, (unsigned*)(ws + OFF_SMX));
  seg_exp_kernel<<<(E_CONST + 255) / 256, 256, 0, stream>>>(
      edges, ws + OFF_LOG, (const unsigned*)(ws + OFF_SMX), ws + OFF_EX,
      ws + OFF_DEN);
  scatter_kernel<<<(E_CONST + 255) / 256, 256, 0, stream>>>(
      edges, ws + OFF_EX, ws + OFF_DEN, na, ws + OFF_NA);
  norm_kernel<<<B_CONST, 512, 0, stream>>>(ws + OFF_NA, out);
}